// GraphClassifier_69561290326172
// MI455X (gfx1250) — compile-verified
//
#include <hip/hip_runtime.h>
#include <hip/hip_bf16.h>
#include <math.h>

#define N_LG   196608
#define D      256
#define NUM_RELS 237
#define NB     128
#define NPG    256      // nodes per graph
#define HS     260      // padded LDS row stride (float) -> banks spread by 4

typedef __attribute__((ext_vector_type(2))) float v2f;
typedef __attribute__((ext_vector_type(8))) float v8f;

__device__ __forceinline__ float sigmoidf_(float x) {
  return 1.0f / (1.0f + __expf(-x));
}

// ---------------------------------------------------------------------------
// 1) segment mean over relation types: one block per relation, deterministic
// ---------------------------------------------------------------------------
__launch_bounds__(256)
__global__ void segmean_kernel(const float* __restrict__ lg_feats,
                               const int*  __restrict__ lg_type,
                               float* __restrict__ means) {
  const int r = blockIdx.x;       // relation id
  const int d = threadIdx.x;      // feature dim
  float sum = 0.0f;
  int   cnt = 0;
  for (int row = 0; row < N_LG; row += 4) {
    const int4 tt = *(const int4*)(lg_type + row);   // uniform -> scalar loads
    if (tt.x == r) { sum += lg_feats[(size_t)(row + 0) * D + d]; ++cnt; }
    if (tt.y == r) { sum += lg_feats[(size_t)(row + 1) * D + d]; ++cnt; }
    if (tt.z == r) { sum += lg_feats[(size_t)(row + 2) * D + d]; ++cnt; }
    if (tt.w == r) { sum += lg_feats[(size_t)(row + 3) * D + d]; ++cnt; }
  }
  means[r * D + d] = (cnt > 0) ? (sum / (float)cnt) : 0.0f;
}

// ---------------------------------------------------------------------------
// 2) h0 = per-graph max over nodes of raw emds
// ---------------------------------------------------------------------------
__launch_bounds__(256)
__global__ void h0max_kernel(const float* __restrict__ emds,
                             float* __restrict__ h0) {
  const int b = blockIdx.x;
  const int d = threadIdx.x;
  const float* p = emds + (size_t)b * NPG * D + d;
  float m = p[0];
  for (int n = 1; n < NPG; ++n) m = fmaxf(m, p[(size_t)n * D]);
  h0[b * D + d] = m;
}

// ---------------------------------------------------------------------------
// 3) GRU scan: 16 graphs per workgroup (one WMMA M-tile), 8 waves.
//    Wave w owns output-gate tiles T = w + 8*j (j=0..5):
//      j=0,1 -> r-gate halves, j=2,3 -> z-gate halves, j=4,5 -> n-gate halves,
//    all at d in {16w..16w+15} U {16w+128..16w+143} -> wave-local elementwise.
//    Per k-step: A operands (h, x_t) from padded LDS, B operands (W rows)
//    streamed from L2, 12x V_WMMA_F32_16X16X4_F32.
// ---------------------------------------------------------------------------
__launch_bounds__(256)
__global__ void gru_scan_kernel(const float* __restrict__ emds,
                                const float* __restrict__ gru_bias,
                                const float* __restrict__ Wih,   // [768,256]
                                const float* __restrict__ Whh,   // [768,256]
                                const float* __restrict__ bih,   // [768]
                                const float* __restrict__ bhh,   // [768]
                                const float* __restrict__ h0,    // [128,256]
                                float* __restrict__ outA, int sA,
                                float* __restrict__ outB, int sB,
                                int t0, int dt, int nsteps) {
  __shared__ float hbuf[16 * HS];
  __shared__ float xbuf[16 * HS];

  const int tid   = threadIdx.x;
  const int wave  = tid >> 5;            // 0..7
  const int lane  = tid & 31;
  const int l15   = lane & 15;
  const int koff  = (lane >> 4) << 1;    // 0 or 2 (f32 WMMA A/B K split)
  const int mrow  = (lane >> 4) << 3;    // C/D: lanes 16..31 hold M+8
  const int gbase = blockIdx.x << 4;     // first graph of this block

  // load h0 tile into LDS
  for (int i = tid; i < 16 * D; i += 256) {
    const int m = i >> 8, d = i & 255;
    hbuf[m * HS + d] = h0[(gbase + m) * D + d];
  }

  // per-lane gate biases (constant over steps)
  float bsum_r[2], bsum_z[2], bi_n[2], bh_n[2];
#pragma unroll
  for (int hf = 0; hf < 2; ++hf) {
    const int n_r = 16 * wave + 128 * hf + l15;
    bsum_r[hf] = bih[n_r]       + bhh[n_r];
    bsum_z[hf] = bih[n_r + 256] + bhh[n_r + 256];
    bi_n[hf]   = bih[n_r + 512];
    bh_n[hf]   = bhh[n_r + 512];
  }

  // weight-row base offsets for this lane's 6 B-tiles
  int wrow[6];
#pragma unroll
  for (int j = 0; j < 6; ++j)
    wrow[j] = (16 * (wave + 8 * j) + l15) * D + koff;

  for (int s = 0; s < nsteps; ++s) {
    const int t = t0 + dt * s;
    // x_t = relu(node[:,t,:] + gru_bias)
    for (int i = tid; i < 16 * D; i += 256) {
      const int m = i >> 8, d = i & 255;
      const float v = emds[((size_t)(gbase + m) * NPG + t) * D + d] + gru_bias[d];
      xbuf[m * HS + d] = v > 0.0f ? v : 0.0f;
    }
    __syncthreads();

    v8f accI[6] = {};   // x @ W_ih^T tiles
    v8f accH[6] = {};   // h @ W_hh^T tiles

    const float* hrow = hbuf + l15 * HS + koff;
    const float* xrow = xbuf + l15 * HS + koff;
#pragma unroll 2
    for (int k0 = 0; k0 < D; k0 += 4) {
      const v2f ah = *(const v2f*)(hrow + k0);
      const v2f ax = *(const v2f*)(xrow + k0);
#pragma unroll
      for (int j = 0; j < 6; ++j) {
        const v2f bi = *(const v2f*)(Wih + wrow[j] + k0);
        const v2f bh = *(const v2f*)(Whh + wrow[j] + k0);
        accI[j] = __builtin_amdgcn_wmma_f32_16x16x4_f32(
            false, ax, false, bi, (short)0, accI[j], false, false);
        accH[j] = __builtin_amdgcn_wmma_f32_16x16x4_f32(
            false, ah, false, bh, (short)0, accH[j], false, false);
      }
    }
    __syncthreads();   // all GEMM reads of hbuf done before updating h

    // wave-local GRU elementwise update, write h_new back to LDS
#pragma unroll
    for (int hf = 0; hf < 2; ++hf) {
      const int d = 16 * wave + 128 * hf + l15;
#pragma unroll
      for (int v = 0; v < 8; ++v) {
        const float r  = sigmoidf_(accI[hf][v]     + accH[hf][v]     + bsum_r[hf]);
        const float z  = sigmoidf_(accI[2 + hf][v] + accH[2 + hf][v] + bsum_z[hf]);
        const float nn = tanhf(accI[4 + hf][v] + bi_n[hf] +
                               r * (accH[4 + hf][v] + bh_n[hf]));
        const int m = mrow + v;
        const float hold = hbuf[m * HS + d];
        hbuf[m * HS + d] = (1.0f - z) * nn + z * hold;
      }
    }
    __syncthreads();

    if (s == sA || s == sB) {
      float* out = (s == sA) ? outA : outB;
      for (int i = tid; i < 16 * D; i += 256) {
        const int m = i >> 8, d = i & 255;
        out[(gbase + m) * D + d] = hbuf[m * HS + d];
      }
    }
  }
}

// ---------------------------------------------------------------------------
// 4) a = relu([fwd,bwd] @ W3^T + b3) for one row per graph
// ---------------------------------------------------------------------------
__launch_bounds__(256)
__global__ void proj_kernel(const float* __restrict__ fwd,
                            const float* __restrict__ bwd,
                            const float* __restrict__ W3,   // [256,512]
                            const float* __restrict__ b3,
                            float* __restrict__ out) {
  __shared__ float xr[2 * D];
  const int b = blockIdx.x;
  const int d = threadIdx.x;
  xr[d]     = fwd[b * D + d];
  xr[D + d] = bwd[b * D + d];
  __syncthreads();
  float acc = b3[d];
  const float* w = W3 + (size_t)d * 2 * D;
  for (int k = 0; k < 2 * D; ++k) acc += xr[k] * w[k];
  out[b * D + d] = acc > 0.0f ? acc : 0.0f;
}

// ---------------------------------------------------------------------------
// 5) scores = ((head + rel - tail) @ W1^T + b1) @ W2^T + b2
// ---------------------------------------------------------------------------
__launch_bounds__(256)
__global__ void score_kernel(const float* __restrict__ a_head,
                             const float* __restrict__ a_tail,
                             const float* __restrict__ means,
                             const int*  __restrict__ rel_labels,
                             const float* __restrict__ W1,
                             const float* __restrict__ b1,
                             const float* __restrict__ W2,
                             const float* __restrict__ b2,
                             float* __restrict__ out) {
  __shared__ float feat[D];
  __shared__ float red[D];
  const int b = blockIdx.x;
  const int d = threadIdx.x;
  const int rl = rel_labels[b];           // rel+1 >= 1 -> always a mean row
  feat[d] = a_head[b * D + d] + means[rl * D + d] - a_tail[b * D + d];
  __syncthreads();
  float t = b1[d];
  const float* w = W1 + (size_t)d * D;
  for (int k = 0; k < D; ++k) t += feat[k] * w[k];
  red[d] = t * W2[d];
  __syncthreads();
  for (int off = 128; off > 0; off >>= 1) {
    if (d < off) red[d] += red[d + off];
    __syncthreads();
  }
  if (d == 0) out[b] = red[0] + b2[0];
}

// ---------------------------------------------------------------------------
extern "C" void kernel_launch(void* const* d_in, const int* in_sizes, int n_in,
                              void* d_out, int out_size, void* d_ws, size_t ws_size,
                              hipStream_t stream) {
  (void)in_sizes; (void)n_in; (void)out_size; (void)ws_size;

  const float* lg_feats   = (const float*)d_in[0];
  const int*   lg_type    = (const int*)  d_in[1];
  const float* emds       = (const float*)d_in[2];
  const int*   rel_labels = (const int*)  d_in[3];
  const float* gru_bias   = (const float*)d_in[4];
  const float* w_ih_f     = (const float*)d_in[5];
  const float* w_hh_f     = (const float*)d_in[6];
  const float* b_ih_f     = (const float*)d_in[7];
  const float* b_hh_f     = (const float*)d_in[8];
  const float* w_ih_b     = (const float*)d_in[9];
  const float* w_hh_b     = (const float*)d_in[10];
  const float* b_ih_b     = (const float*)d_in[11];
  const float* b_hh_b     = (const float*)d_in[12];
  const float* W3         = (const float*)d_in[13];
  const float* b3         = (const float*)d_in[14];
  const float* W1         = (const float*)d_in[15];
  const float* b1         = (const float*)d_in[16];
  const float* W2         = (const float*)d_in[17];
  const float* b2         = (const float*)d_in[18];
  float* outp = (float*)d_out;

  float* ws     = (float*)d_ws;
  float* means  = ws;                         // 237*256
  float* h0     = means + NUM_RELS * D;       // 128*256
  float* fwd0   = h0    + NB * D;
  float* fwd1   = fwd0  + NB * D;
  float* bwd0   = fwd1  + NB * D;
  float* bwd1   = bwd0  + NB * D;
  float* ahead  = bwd1  + NB * D;
  float* atail  = ahead + NB * D;

  segmean_kernel<<<NUM_RELS, 256, 0, stream>>>(lg_feats, lg_type, means);
  h0max_kernel<<<NB, 256, 0, stream>>>(emds, h0);

  // forward GRU: only steps 0 and 1 are consumed downstream
  gru_scan_kernel<<<NB / 16, 256, 0, stream>>>(
      emds, gru_bias, w_ih_f, w_hh_f, b_ih_f, b_hh_f, h0,
      fwd0, /*sA=*/0, fwd1, /*sB=*/1, /*t0=*/0, /*dt=*/1, /*nsteps=*/2);
  // backward GRU: full 256-step scan; bwd[1]=after 255 steps, bwd[0]=after 256
  gru_scan_kernel<<<NB / 16, 256, 0, stream>>>(
      emds, gru_bias, w_ih_b, w_hh_b, b_ih_b, b_hh_b, h0,
      bwd0, /*sA=*/255, bwd1, /*sB=*/254, /*t0=*/255, /*dt=*/-1, /*nsteps=*/256);

  proj_kernel<<<NB, 256, 0, stream>>>(fwd0, bwd0, W3, b3, ahead);
  proj_kernel<<<NB, 256, 0, stream>>>(fwd1, bwd1, W3, b3, atail);

  score_kernel<<<NB, 256, 0, stream>>>(ahead, atail, means, rel_labels,
                                       W1, b1, W2, b2, outp);
}